// U_S_encoder_5583457485490
// MI455X (gfx1250) — compile-verified
//
#include <hip/hip_runtime.h>
#include <hip/hip_bf16.h>
#include <math.h>

#define N_NODES 10000
#define N_EDGES 640000
#define D_IN    128
#define D_HID   128
#define D_LAT   64

typedef __attribute__((ext_vector_type(2))) float v2f;
typedef __attribute__((ext_vector_type(8))) float v8f;

// ---------------------------------------------------------------- utilities

__global__ __launch_bounds__(256) void fill_kernel(float* __restrict__ p, float v, int n) {
    int i = blockIdx.x * blockDim.x + threadIdx.x;
    if (i < n) p[i] = v;
}

__global__ __launch_bounds__(256) void degree_kernel(const int* __restrict__ dst,
                                                     float* __restrict__ deg, int n_edges) {
    int e = blockIdx.x * blockDim.x + threadIdx.x;
    if (e < n_edges) unsafeAtomicAdd(&deg[dst[e]], 1.0f);
}

__global__ __launch_bounds__(256) void rsqrt_kernel(float* __restrict__ d, int n) {
    int i = blockIdx.x * blockDim.x + threadIdx.x;
    if (i < n) d[i] = rsqrtf(d[i]);
}

// ------------------------------------------------- WMMA GEMM 1: H0 = X @ W1
// One wave computes one 16x16 tile of [10000 x 128]; 8 waves/block = 16-row strip.
__global__ __launch_bounds__(256) void gemm_x_w1(const float* __restrict__ X,
                                                 const float* __restrict__ W,
                                                 float* __restrict__ H) {
    const int lane = threadIdx.x & 31;
    const int wave = threadIdx.x >> 5;
    const int row0 = blockIdx.x * 16;       // 625 blocks
    const int col0 = wave * 16;             // 8 waves -> 128 cols
    const int m  = lane & 15;
    const int kk = (lane >> 4) << 1;        // lanes 0-15 -> K0/K1, 16-31 -> K2/K3

    v8f c = {};
    const float* xrow = X + (row0 + m) * D_IN;
#pragma unroll 4
    for (int k = 0; k < D_IN; k += 4) {
        v2f a, b;
        a.x = xrow[k + kk];
        a.y = xrow[k + kk + 1];
        b.x = W[(k + kk) * D_HID + col0 + m];
        b.y = W[(k + kk + 1) * D_HID + col0 + m];
        c = __builtin_amdgcn_wmma_f32_16x16x4_f32(false, a, false, b,
                                                  (short)0, c, false, false);
    }
    const int rbase = (lane >> 4) << 3;     // 0 or 8
#pragma unroll
    for (int r = 0; r < 8; ++r)
        H[(row0 + rbase + r) * D_HID + col0 + m] = c[r];
}

// --------------------------------- WMMA GEMM 2: T[:,0:64]=H@Wmu, T[:,64:128]=H@Wls
__global__ __launch_bounds__(256) void gemm_h_w2(const float* __restrict__ H,
                                                 const float* __restrict__ Wmu,
                                                 const float* __restrict__ Wls,
                                                 float* __restrict__ T) {
    const int lane = threadIdx.x & 31;
    const int wave = threadIdx.x >> 5;
    const int row0 = blockIdx.x * 16;       // 625 blocks
    const int is_ls = wave >> 2;            // waves 0-3: mu, 4-7: ls
    const int col0 = (wave & 3) * 16;       // 4 waves -> 64 cols
    const float* W = is_ls ? Wls : Wmu;
    const int m  = lane & 15;
    const int kk = (lane >> 4) << 1;

    v8f c = {};
    const float* hrow = H + (row0 + m) * D_HID;
#pragma unroll 4
    for (int k = 0; k < D_HID; k += 4) {
        v2f a, b;
        a.x = hrow[k + kk];
        a.y = hrow[k + kk + 1];
        b.x = W[(k + kk) * D_LAT + col0 + m];
        b.y = W[(k + kk + 1) * D_LAT + col0 + m];
        c = __builtin_amdgcn_wmma_f32_16x16x4_f32(false, a, false, b,
                                                  (short)0, c, false, false);
    }
    const int outc  = is_ls * D_LAT + col0 + m;   // packed [N][128]: mu | ls
    const int rbase = (lane >> 4) << 3;
#pragma unroll
    for (int r = 0; r < 8; ++r)
        T[(row0 + rbase + r) * 128 + outc] = c[r];
}

// ------------------------------------- edge scatter: agg[dst] += feat[src]*en
// 32 threads per edge, 4 features each (float4 gather, native f32 atomics).
__global__ __launch_bounds__(256) void scatter_kernel(const float* __restrict__ feat,
                                                      const int* __restrict__ srcI,
                                                      const int* __restrict__ dstI,
                                                      const float* __restrict__ dis,
                                                      float* __restrict__ agg) {
    unsigned tid = blockIdx.x * blockDim.x + threadIdx.x;
    unsigned e = tid >> 5;
    if (e >= N_EDGES) return;
    int f = (int)(tid & 31) * 4;
    int s = srcI[e];
    int d = dstI[e];
    float en = dis[s] * dis[d];
    const float4 v = *(const float4*)(feat + (size_t)s * 128 + f);
    float* o = agg + (size_t)d * 128 + f;
    unsafeAtomicAdd(o + 0, v.x * en);
    unsafeAtomicAdd(o + 1, v.y * en);
    unsafeAtomicAdd(o + 2, v.z * en);
    unsafeAtomicAdd(o + 3, v.w * en);
}

// ------------------------------------ layer-1 epilogue: h = relu(agg + h0*sn + b1)
__global__ __launch_bounds__(256) void finalize1_kernel(const float* __restrict__ h0,
                                                        float* __restrict__ agg,
                                                        const float* __restrict__ dis,
                                                        const float* __restrict__ b1) {
    int i = blockIdx.x * blockDim.x + threadIdx.x;
    if (i >= N_NODES * D_HID) return;
    int node = i >> 7;
    int f = i & 127;
    float sn = dis[node];
    sn *= sn;
    float v = agg[i] + h0[i] * sn + b1[f];
    agg[i] = fmaxf(v, 0.0f);
}

// -------------------- layer-2 epilogue: mu + bias; softplus(logstd + bias) -> out
__global__ __launch_bounds__(256) void finalize2_kernel(const float* __restrict__ t,
                                                        const float* __restrict__ agg,
                                                        const float* __restrict__ dis,
                                                        const float* __restrict__ bmu,
                                                        const float* __restrict__ bls,
                                                        float* __restrict__ out) {
    int i = blockIdx.x * blockDim.x + threadIdx.x;
    if (i >= N_NODES * D_LAT) return;
    int node = i >> 6;
    int f = i & 63;
    float sn = dis[node];
    sn *= sn;
    size_t base = (size_t)node * 128;
    float mu = agg[base + f] + t[base + f] * sn + bmu[f];
    float ls = agg[base + 64 + f] + t[base + 64 + f] * sn + bls[f];
    out[i] = mu;
    // numerically stable softplus
    float sp = (ls > 20.0f) ? ls : log1pf(expf(ls));
    out[N_NODES * D_LAT + i] = sp;
}

// ---------------------------------------------------------------- launcher

extern "C" void kernel_launch(void* const* d_in, const int* in_sizes, int n_in,
                              void* d_out, int out_size, void* d_ws, size_t ws_size,
                              hipStream_t stream) {
    const float* X   = (const float*)d_in[0];
    const int*   ei  = (const int*)d_in[1];       // [2][E]: row0 = src, row1 = dst
    const float* W1  = (const float*)d_in[2];
    const float* b1  = (const float*)d_in[3];
    const float* Wmu = (const float*)d_in[4];
    const float* bmu = (const float*)d_in[5];
    const float* Wls = (const float*)d_in[6];
    const float* bls = (const float*)d_in[7];
    float* out = (float*)d_out;

    const int* srcI = ei;
    const int* dstI = ei + N_EDGES;

    // workspace layout (floats): dis[10240] | bufA[N*128] | bufB[N*128]  ~= 10.3 MB
    float* dis  = (float*)d_ws;
    float* bufA = dis + 10240;
    float* bufB = bufA + (size_t)N_NODES * 128;

    const int TB = 256;
    const int gridNF   = (N_NODES * 128 + TB - 1) / TB;       // elementwise over N*128
    const int gridNL   = (N_NODES * D_LAT + TB - 1) / TB;     // elementwise over N*64
    const int gridE    = (N_EDGES + TB - 1) / TB;             // per-edge
    const int gridScat = (N_EDGES * 32) / TB;                 // 32 threads/edge
    const int gridGemm = N_NODES / 16;                        // 625 row-tiles

    // 1) degree -> dis = rsqrt(deg + 1)
    fill_kernel<<<(N_NODES + TB - 1) / TB, TB, 0, stream>>>(dis, 1.0f, N_NODES);
    degree_kernel<<<gridE, TB, 0, stream>>>(dstI, dis, N_EDGES);
    rsqrt_kernel<<<(N_NODES + TB - 1) / TB, TB, 0, stream>>>(dis, N_NODES);

    // 2) H0 = X @ W1   (WMMA f32)
    gemm_x_w1<<<gridGemm, TB, 0, stream>>>(X, W1, bufA);

    // 3) agg = segment_sum(H0[src] * en, dst)
    fill_kernel<<<gridNF, TB, 0, stream>>>(bufB, 0.0f, N_NODES * 128);
    scatter_kernel<<<gridScat, TB, 0, stream>>>(bufA, srcI, dstI, dis, bufB);

    // 4) h = relu(agg + H0*sn + b1)   (in place in bufB)
    finalize1_kernel<<<gridNF, TB, 0, stream>>>(bufA, bufB, dis, b1);

    // 5) T = [h@Wmu | h@Wls]   (WMMA f32, packed [N][128])
    gemm_h_w2<<<gridGemm, TB, 0, stream>>>(bufB, Wmu, Wls, bufA);

    // 6) agg2 = segment_sum(T[src] * en, dst)   (reuse bufB)
    fill_kernel<<<gridNF, TB, 0, stream>>>(bufB, 0.0f, N_NODES * 128);
    scatter_kernel<<<gridScat, TB, 0, stream>>>(bufA, srcI, dstI, dis, bufB);

    // 7) mu / softplus(logstd) -> out
    finalize2_kernel<<<gridNL, TB, 0, stream>>>(bufA, bufB, dis, bmu, bls, out);
}